// IOULoss_70514773066125
// MI455X (gfx1250) — compile-verified
//
#include <hip/hip_runtime.h>

typedef __attribute__((ext_vector_type(2))) float v2f;
typedef __attribute__((ext_vector_type(8))) float v8f;

// Sum 32 lane-private floats across the wave using V_WMMA_F32_16X16X4_F32.
// A (16x4): lane L(0..15) -> A[M=L][K=0]=x, A[M=L][K=1]=0
//           lane L(16..31)-> A[M=L-16][K=2]=x, [K=3]=0
// B (4x16) = all ones  =>  D[M][N] = x[M] + x[M+16]  (independent of N).
// Per-lane sum of the 8 D VGPRs gives sum(M=0..7) on lanes 0..15 and
// sum(M=8..15) on lanes 16..31; shfl_xor(16) completes the total.
// Caller must guarantee the whole wave executes this (EXEC all ones).
__device__ __forceinline__ float wave32_sum_wmma(float x) {
    v2f a; a[0] = x;    a[1] = 0.0f;
    v2f b; b[0] = 1.0f; b[1] = 1.0f;
    v8f c = {};
    c = __builtin_amdgcn_wmma_f32_16x16x4_f32(
        /*neg_a=*/false, a, /*neg_b=*/false, b,
        /*c_mod=*/(short)0, c, /*reuse_a=*/false, /*reuse_b=*/false);
    float r = c[0] + c[1] + c[2] + c[3] + c[4] + c[5] + c[6] + c[7];
    r += __shfl_xor(r, 16, 32);
    return r;
}

// Layer-4 IoU log-loss partial sums. Grid: 256 blocks x 256 threads.
// blockIdx -> (b = blk>>4, n-chunk = (blk&15)*256). Each thread owns one n,
// loops t over the 96 gt intervals held in LDS (uniform broadcast reads).
// Accumulates log2(den)-log2(num) for valid pairs; block sum -> ws[blk].
__global__ void __launch_bounds__(256)
iou_partial_kernel(const float* __restrict__ pred,
                   const float* __restrict__ gt,
                   float* __restrict__ ws) {
    constexpr int B = 16, N = 4096, T = 96;
    constexpr float EPS = 1e-6f;

    const int tid = threadIdx.x;
    const int b  = blockIdx.x >> 4;
    const int n  = ((blockIdx.x & 15) << 8) + tid;

    __shared__ float2 gt_s[T];
    const float2* gt2 = (const float2*)(gt + (size_t)4 * B * T * 2) + (size_t)b * T;
    if (tid < T) gt_s[tid] = gt2[tid];
    __syncthreads();

    const float2* pred2 = (const float2*)(pred + (size_t)4 * B * N * 2) + (size_t)b * N;
    const float2 pe = pred2[n];
    const float s = pe.x, e = pe.y;

    float acc = 0.0f;
#pragma unroll 4
    for (int t = 0; t < T; ++t) {
        const float a  = gt_s[t].x;
        const float bb = gt_s[t].y;
        // mutually exclusive containment cases (bitwise & keeps it branchless)
        const bool c0 = (s < a) & (e > bb);               // pred covers gt
        const bool c1 = (s < a) & (e < bb) & (e > a);     // left overlap
        const bool c2 = (s > a) & (s < bb) & (e > bb);    // right overlap
        const float num = c0 ? (bb - a) : (c1 ? (e - a)  : (bb - s));
        const float den = c0 ? (e - s)  : (c1 ? (bb - s) : (e - a));
        // iou < EPS (incl. no-overlap iou==0) -> clamped to 1.0 -> zero loss
        const bool valid = (c0 | c1 | c2) & (num >= EPS * den);
        const float term = __log2f(den) - __log2f(num);   // = -log2(iou)
        acc += valid ? term : 0.0f;                        // NaN masked by select
    }

    __shared__ float red[256];
    red[tid] = acc;
    __syncthreads();
    if (tid < 128) red[tid] += red[tid + 128];
    __syncthreads();
    if (tid < 64)  red[tid] += red[tid + 64];
    __syncthreads();
    if (tid < 32) {                       // wave 0 only: EXEC all ones for WMMA
        const float v = red[tid] + red[tid + 32];
        const float tot = wave32_sum_wmma(v);
        if (tid == 0) ws[blockIdx.x] = tot;
    }
}

// Final deterministic 256 -> 1 reduction (single wave), WMMA reduce, then
// convert log2-domain sum to natural log and divide by batch_size=16.
__global__ void __launch_bounds__(32)
iou_final_kernel(const float* __restrict__ ws, float* __restrict__ out) {
    const int tid = threadIdx.x;
    float v = 0.0f;
#pragma unroll
    for (int i = 0; i < 8; ++i) v += ws[tid + 32 * i];
    const float tot = wave32_sum_wmma(v);
    if (tid == 0) out[0] = tot * (0.69314718055994530942f * 0.0625f); // ln2 / 16
}

extern "C" void kernel_launch(void* const* d_in, const int* in_sizes, int n_in,
                              void* d_out, int out_size, void* d_ws, size_t ws_size,
                              hipStream_t stream) {
    (void)in_sizes; (void)n_in; (void)out_size; (void)ws_size;
    const float* pred = (const float*)d_in[0];   // (5,16,4096,2) f32
    const float* gt   = (const float*)d_in[1];   // (5,16,96,2)  f32
    float* out = (float*)d_out;                  // scalar f32
    float* ws  = (float*)d_ws;                   // 256 partial sums

    iou_partial_kernel<<<256, 256, 0, stream>>>(pred, gt, ws);
    iou_final_kernel<<<1, 32, 0, stream>>>(ws, out);
}